// sentence_genaration_15135464751216
// MI455X (gfx1250) — compile-verified
//
#include <hip/hip_runtime.h>
#include <hip/hip_bf16.h>

#define BB 4
#define LL 8192
#define DD 768
#define MAXS 100

typedef __attribute__((ext_vector_type(2))) float v2f;
typedef __attribute__((ext_vector_type(8))) float v8f;

// ---------------------------------------------------------------------------
// Phase 1: contiguous-span segment max pooling.
// grid = (B*MAXS, 2), block = 128 threads; each block owns one (batch, segment)
// pair and half of the 768 columns (384 cols = 128 threads x 3 cols).
// Sentence ids are sorted per row, so segment j (id j+1) is the span
// [lower_bound(j+1), lower_bound(j+2)).  bb = ids[b][L-1] (sorted).
// ---------------------------------------------------------------------------
__global__ void __launch_bounds__(128)
seg_max_kernel(const float* __restrict__ wf,
               const int* __restrict__ ids,
               float* __restrict__ pooled) {
    const int bj = blockIdx.x;            // 0 .. B*MAXS-1
    const int b  = bj / MAXS;
    const int j  = bj % MAXS;             // segment j pools sentence id j+1
    const int colBase = blockIdx.y * (DD / 2);   // 0 or 384
    const int tid = threadIdx.x;          // 0..127

    const int* idrow = ids + b * LL;
    const int bb = idrow[LL - 1];         // max id (row is sorted ascending)

    // branch-free-ish binary searches (uniform across the block -> scalarized)
    int s, e;
    {
        int lo = 0, hi = LL;              // lower_bound(j+1)
        while (lo < hi) { int mid = (lo + hi) >> 1; if (idrow[mid] < j + 1) lo = mid + 1; else hi = mid; }
        s = lo;
        lo = s; hi = LL;                  // lower_bound(j+2), monotone: start at s
        while (lo < hi) { int mid = (lo + hi) >> 1; if (idrow[mid] < j + 2) lo = mid + 1; else hi = mid; }
        e = lo;
    }

    const float NEG_INF = -__builtin_inff();   // segment_max identity (matches jax)
    float m0 = NEG_INF, m1 = NEG_INF, m2 = NEG_INF;

    const float* base = wf + (size_t)b * LL * DD + colBase + tid;
    for (int t = s; t < e; ++t) {
        const float* row = base + (size_t)t * DD;
        // hint the streaming pattern to the prefetcher (global_prefetch_b8)
        __builtin_prefetch(base + (size_t)(t + 8) * DD, 0, 0);
        m0 = fmaxf(m0, row[0]);
        m1 = fmaxf(m1, row[128]);
        m2 = fmaxf(m2, row[256]);
    }

    if (j >= bb) { m0 = 0.0f; m1 = 0.0f; m2 = 0.0f; }   // invalid sentences -> 0

    float* prow = pooled + (size_t)bj * DD + colBase + tid;
    prow[0]   = m0;
    prow[128] = m1;
    prow[256] = m2;
}

// ---------------------------------------------------------------------------
// Phase 2: out[r, n] = sum_k pooled[r, k] * W2[n, k] + b2[n]
// f32 WMMA 16x16x4 keeps full f32 precision (GEMM is only ~472 MFLOP; the
// kernel is bandwidth-bound overall, so no reason to drop to bf16).
// grid = (48/8, 25), block = 256 (8 waves); each wave owns one 16x16 tile.
//
// A fragment (16x4 f32): lane l<16 -> A[M=l,    K=0..1]  (float2)
//                        lane l>=16 -> A[M=l-16, K=2..3]  (float2)
// B fragment (4x16 f32): lane l<16 -> B[K=0..1, N=l]     = W2[nb+l][k+0..1]
//                        lane l>=16 -> B[K=2..3, N=l-16] = W2[nb+l-16][k+2..3]
// D (16x16 f32, 8 VGPRs): vgpr r, lanes 0-15 -> (M=r, N=lane),
//                                  lanes 16-31 -> (M=8+r, N=lane-16)
// ---------------------------------------------------------------------------
__global__ void __launch_bounds__(256)
wmma_gemm_kernel(const float* __restrict__ apool,
                 const float* __restrict__ w2,
                 const float* __restrict__ bias,
                 float* __restrict__ out) {
    const int lane = threadIdx.x & 31;
    const int wave = threadIdx.x >> 5;
    const int ntile = blockIdx.x * 8 + wave;   // 0..47  (N = 768 = 48*16)
    const int mtile = blockIdx.y;              // 0..24  (M = 400 = 25*16)
    const int mbase = mtile * 16;
    const int nbase = ntile * 16;
    const int half  = lane >> 4;               // 0: K=0..1 / M rows 0..7 half
    const int l16   = lane & 15;

    const float* arow = apool + (size_t)(mbase + l16) * DD + 2 * half;
    const float* brow = w2    + (size_t)(nbase + l16) * DD + 2 * half;

    v8f acc = {};
#pragma unroll 8
    for (int k = 0; k < DD; k += 4) {
        v2f afrag = *(const v2f*)(arow + k);
        v2f bfrag = *(const v2f*)(brow + k);
        acc = __builtin_amdgcn_wmma_f32_16x16x4_f32(
            /*neg_a=*/false, afrag,
            /*neg_b=*/false, bfrag,
            /*c_mod=*/(short)0, acc,
            /*reuse_a=*/false, /*reuse_b=*/false);
    }

    const int col = nbase + l16;
    const float bv = bias[col];
    float* obase = out + (size_t)(mbase + 8 * half) * DD + col;
#pragma unroll
    for (int r = 0; r < 8; ++r) {
        obase[(size_t)r * DD] = acc[r] + bv;
    }
}

// ---------------------------------------------------------------------------
extern "C" void kernel_launch(void* const* d_in, const int* in_sizes, int n_in,
                              void* d_out, int out_size, void* d_ws, size_t ws_size,
                              hipStream_t stream) {
    const float* word_feature  = (const float*)d_in[0];   // [4, 8192, 768] f32
    const int*   sentence_mask = (const int*)  d_in[1];   // [4, 64, 128] int32 (== [4, 8192])
    const float* w2            = (const float*)d_in[2];   // [768, 768] f32
    const float* b2            = (const float*)d_in[3];   // [768] f32
    float*       out           = (float*)d_out;           // [4, 100, 768] f32

    float* pooled = (float*)d_ws;                         // [400, 768] f32 = 1.2 MB

    dim3 g1(BB * MAXS, 2);
    seg_max_kernel<<<g1, 128, 0, stream>>>(word_feature, sentence_mask, pooled);

    dim3 g2(48 / 8, (BB * MAXS) / 16);                    // (6, 25)
    wmma_gemm_kernel<<<g2, 256, 0, stream>>>(pooled, w2, b2, out);
}